// Sinkhorn_87960930222185
// MI455X (gfx1250) — compile-verified
//
#include <hip/hip_runtime.h>
#include <math.h>

typedef __attribute__((ext_vector_type(2))) float v2f;
typedef __attribute__((ext_vector_type(8))) float v8f;

#define B_    16
#define C_    768
#define HW_   4096
#define D_    256
#define M_    32
#define EPS_  0.05f
#define NORM_ (-8.3255517f)   /* -log(32 + 4096) */
#define NEPS_ 1e-12f

/* workspace layout (float offsets) */
#define OFF_WHAT  0                      /* 32*256   = 8192   */
#define OFF_Z     8192                   /* 16*32*4096 = 2097152 */
#define OFF_U     (8192 + 2097152)       /* 16*32    = 512    */
#define OFF_V     (OFF_U + 512)          /* 16*4096  = 65536  */
#define OFF_RINV  (OFF_V + 65536)        /* 16*4096  = 65536  */

static __device__ __forceinline__ v8f wmma4(v2f a, v2f b, v8f c) {
  /* D = A(16x4 f32) * B(4x16 f32) + C(16x16 f32) */
  return __builtin_amdgcn_wmma_f32_16x16x4_f32(false, a, false, b, (short)0, c,
                                               false, false);
}

/* ---- normalize cluster weights: what[m][d] = v[m][d] / max(||v_m||, eps) ---- */
__global__ void k_norm_v(const float* __restrict__ v, float* __restrict__ what) {
  int m = threadIdx.x;
  if (m < M_) {
    float s = 0.f;
    for (int d = 0; d < D_; ++d) { float t = v[m * D_ + d]; s += t * t; }
    float inv = 1.0f / fmaxf(sqrtf(s), NEPS_);
    for (int d = 0; d < D_; ++d) what[m * D_ + d] = v[m * D_ + d] * inv;
  }
}

__global__ void k_zero(float* __restrict__ p, int n) {
  int i = blockIdx.x * blockDim.x + threadIdx.x;
  if (i < n) p[i] = 0.f;
}

/* ---- fused: 1x1-conv GEMM + bias + x_proj store + row L2 norm + scores ----
   block = (batch b, 16-row tile n0); 256 threads = 8 waves.
   Each wave owns 32 output columns (two 16x16 WMMA tiles), K = 768.      */
__global__ void __launch_bounds__(256) k_proj(
    const float* __restrict__ x, const float* __restrict__ w,
    const float* __restrict__ bias, const float* __restrict__ what,
    float* __restrict__ xp, float* __restrict__ Z, float* __restrict__ rinvg) {
  __shared__ float sA[C_ * 16];      /* [c][n] : conflict-free A reads */
  __shared__ float sRed[8][16];
  __shared__ float sRinv[16];

  const int tid = threadIdx.x;
  const int b = blockIdx.x >> 8;
  const int n0 = (blockIdx.x & 255) * 16;
  const int wave = tid >> 5;
  const int lane = tid & 31;
  const int laneLo = lane & 15;
  const int laneHi = lane >> 4;

  { /* stage A tile: sA[c*16+n] = x[b][c][n0+n] (coalesced 64B rows) */
    const float* xb = x + (size_t)b * C_ * HW_ + n0;
    for (int i = tid; i < C_ * 16; i += 256) {
      int c = i >> 4, n = i & 15;
      sA[i] = xb[(size_t)c * HW_ + n];
    }
  }
  __syncthreads();

  const int d0 = wave * 32;
  v8f acc0 = {}; v8f acc1 = {};
  const float* w0 = w + (size_t)(d0 + laneLo) * C_;
  const float* w1 = w + (size_t)(d0 + 16 + laneLo) * C_;
#pragma unroll 4
  for (int k = 0; k < C_; k += 4) {
    int kk = k + 2 * laneHi;
    v2f a; a.x = sA[kk * 16 + laneLo]; a.y = sA[(kk + 1) * 16 + laneLo];
    v2f b0 = *(const v2f*)(w0 + kk);
    v2f b1 = *(const v2f*)(w1 + kk);
    acc0 = wmma4(a, b0, acc0);
    acc1 = wmma4(a, b1, acc1);
  }

  const float bi0 = bias[d0 + laneLo];
  const float bi1 = bias[d0 + 16 + laneLo];
#pragma unroll
  for (int r = 0; r < 8; ++r) { acc0[r] += bi0; acc1[r] += bi1; }

  { /* store x_proj[b][n][d] */
    float* xpb = xp + (size_t)b * HW_ * D_ + (size_t)n0 * D_;
#pragma unroll
    for (int r = 0; r < 8; ++r) {
      int row = r + 8 * laneHi;
      xpb[(size_t)row * D_ + d0 + laneLo] = acc0[r];
      xpb[(size_t)row * D_ + d0 + 16 + laneLo] = acc1[r];
    }
  }

  /* deterministic per-row sum of squares: xor-shuffle within 16-lane halves */
#pragma unroll
  for (int r = 0; r < 8; ++r) {
    float val = acc0[r] * acc0[r] + acc1[r] * acc1[r];
    for (int off = 1; off < 16; off <<= 1) val += __shfl_xor(val, off, 32);
    if (laneLo == 0) sRed[wave][r + 8 * laneHi] = val;
  }
  __syncthreads();  /* also: all waves done reading sA */

  if (tid < 16) {
    float s = 0.f;
    for (int wv = 0; wv < 8; ++wv) s += sRed[wv][tid];
    float inv = 1.0f / fmaxf(sqrtf(s), NEPS_);
    sRinv[tid] = inv;
    rinvg[(size_t)b * HW_ + n0 + tid] = inv;
  }

  /* re-stage tile (alias sA): sP[d*16 + n] for the scores GEMM */
  float* sP = sA;
#pragma unroll
  for (int r = 0; r < 8; ++r) {
    int row = r + 8 * laneHi;
    sP[(d0 + laneLo) * 16 + row] = acc0[r];
    sP[(d0 + 16 + laneLo) * 16 + row] = acc1[r];
  }
  __syncthreads();

  /* scores: waves 0,1 each compute a 16(n) x 16(m) tile over K = 256,
     then write Z[b][m][n] = score * rinv[n] / eps                      */
  if (wave < 2) {
    const int m0 = wave * 16;
    v8f cs = {};
    const float* wh = what + (size_t)(m0 + laneLo) * D_;
#pragma unroll 4
    for (int k = 0; k < D_; k += 4) {
      int kk = k + 2 * laneHi;
      v2f a; a.x = sP[kk * 16 + laneLo]; a.y = sP[(kk + 1) * 16 + laneLo];
      v2f bb = *(const v2f*)(wh + kk);
      cs = wmma4(a, bb, cs);
    }
    float* Zb = Z + (size_t)b * (M_ * HW_) + (size_t)(m0 + laneLo) * HW_ + n0;
    const float invEps = 1.0f / EPS_;
#pragma unroll
    for (int r = 0; r < 8; ++r) {
      int row = r + 8 * laneHi;
      Zb[row] = cs[r] * sRinv[row] * invEps;
    }
  }
}

/* ---- Sinkhorn u-update: u[b][m] = norm - LSE_n(Z[b,m,n] + v[n]) ---- */
__global__ void __launch_bounds__(256) k_u(const float* __restrict__ Z,
                                           const float* __restrict__ vdual,
                                           float* __restrict__ u) {
  __shared__ float sb[256];
  const int b = blockIdx.x >> 5;
  const int m = blockIdx.x & 31;
  const float* zrow = Z + (size_t)b * (M_ * HW_) + (size_t)m * HW_;
  const float* vv = vdual + (size_t)b * HW_;
  const int tid = threadIdx.x;

  float mx = -1e30f;
  for (int n = tid; n < HW_; n += 256) mx = fmaxf(mx, zrow[n] + vv[n]);
  sb[tid] = mx; __syncthreads();
  for (int s = 128; s > 0; s >>= 1) {
    if (tid < s) sb[tid] = fmaxf(sb[tid], sb[tid + s]);
    __syncthreads();
  }
  const float gmax = sb[0]; __syncthreads();

  float sm = 0.f;
  for (int n = tid; n < HW_; n += 256) sm += expf(zrow[n] + vv[n] - gmax);
  sb[tid] = sm; __syncthreads();
  for (int s = 128; s > 0; s >>= 1) {
    if (tid < s) sb[tid] += sb[tid + s];
    __syncthreads();
  }
  if (tid == 0) u[b * M_ + m] = NORM_ - (gmax + logf(sb[0]));
}

/* ---- Sinkhorn v-update: v[b][n] = norm - LSE_m(Z[b,m,n] + u[m]) ---- */
__global__ void __launch_bounds__(256) k_vdual(const float* __restrict__ Z,
                                               const float* __restrict__ u,
                                               float* __restrict__ vdual) {
  __shared__ float sU[M_];
  const int idx = blockIdx.x * 256 + threadIdx.x;
  const int b = idx >> 12;
  const int n = idx & 4095;
  if (threadIdx.x < M_) sU[threadIdx.x] = u[b * M_ + threadIdx.x];
  __syncthreads();
  const float* zb = Z + (size_t)b * (M_ * HW_) + n;
  float mx = -1e30f;
#pragma unroll
  for (int m = 0; m < M_; ++m) mx = fmaxf(mx, zb[(size_t)m * HW_] + sU[m]);
  float sm = 0.f;
#pragma unroll
  for (int m = 0; m < M_; ++m) sm += expf(zb[(size_t)m * HW_] + sU[m] - mx);
  vdual[idx] = NORM_ - (mx + logf(sm));
}

/* ---- v_tilde[b][m][d] = sum_n exp(Z+u+v-norm)[m][n] * (x_proj[n][d]*rinv[n])
   one block (8 waves) per 16x16 (m,d) tile; waves split K=4096 into 512-wide
   slices, accumulate via f32 WMMA, then deterministic LDS reduction.     ---- */
__global__ void __launch_bounds__(256) k_vtilde(
    const float* __restrict__ Z, const float* __restrict__ u,
    const float* __restrict__ vdual, const float* __restrict__ xp,
    const float* __restrict__ rinv, float* __restrict__ vt) {
  __shared__ float sAcc[8][32][9];   /* pad to 9: conflict-free */
  const int tid = threadIdx.x;
  const int wave = tid >> 5;
  const int lane = tid & 31;
  const int laneLo = lane & 15;
  const int laneHi = lane >> 4;

  const int b = blockIdx.x >> 5;       /* 32 tiles per batch */
  const int rem = blockIdx.x & 31;
  const int m0 = (rem >> 4) * 16;
  const int d0 = (rem & 15) * 16;

  const int k_beg = wave * (HW_ / 8);  /* 512-wide K slice per wave */
  const int k_end = k_beg + (HW_ / 8);

  const float* zrow = Z + (size_t)b * (M_ * HW_) + (size_t)(m0 + laneLo) * HW_;
  const float um = u[b * M_ + m0 + laneLo];
  const float* vv = vdual + (size_t)b * HW_;
  const float* xpb = xp + (size_t)b * HW_ * D_ + d0 + laneLo;
  const float* rv = rinv + (size_t)b * HW_;

  v8f acc = {};
  for (int k = k_beg; k < k_end; k += 4) {
    int kk = k + 2 * laneHi;
    v2f z2 = *(const v2f*)(zrow + kk);
    v2f vw = *(const v2f*)(vv + kk);
    v2f a;
    a.x = expf(z2.x + um + vw.x - NORM_);
    a.y = expf(z2.y + um + vw.y - NORM_);
    v2f r2 = *(const v2f*)(rv + kk);
    v2f bb;
    bb.x = xpb[(size_t)kk * D_] * r2.x;
    bb.y = xpb[(size_t)(kk + 1) * D_] * r2.y;
    acc = wmma4(a, bb, acc);
  }

#pragma unroll
  for (int r = 0; r < 8; ++r) sAcc[wave][lane][r] = acc[r];
  __syncthreads();

  if (wave == 0) {
    float* out = vt + (size_t)b * M_ * D_ + (size_t)m0 * D_ + d0 + laneLo;
#pragma unroll
    for (int r = 0; r < 8; ++r) {
      float s = 0.f;
      for (int w2 = 0; w2 < 8; ++w2) s += sAcc[w2][lane][r];
      int row = r + 8 * laneHi;
      out[(size_t)row * D_] = s;
    }
  }
}

extern "C" void kernel_launch(void* const* d_in, const int* in_sizes, int n_in,
                              void* d_out, int out_size, void* d_ws, size_t ws_size,
                              hipStream_t stream) {
  const float* x      = (const float*)d_in[0];
  const float* conv_w = (const float*)d_in[1];
  const float* conv_b = (const float*)d_in[2];
  const float* v      = (const float*)d_in[3];

  float* out_vt = (float*)d_out;                       /* [16,32,256]  */
  float* out_xp = out_vt + (size_t)B_ * M_ * D_;       /* [16,4096,256]*/

  float* ws   = (float*)d_ws;
  float* what = ws + OFF_WHAT;
  float* Z    = ws + OFF_Z;
  float* u    = ws + OFF_U;
  float* vd   = ws + OFF_V;
  float* rinv = ws + OFF_RINV;

  k_norm_v<<<1, 256, 0, stream>>>(v, what);
  k_zero<<<(B_ * HW_ + 255) / 256, 256, 0, stream>>>(vd, B_ * HW_);
  k_proj<<<B_ * (HW_ / 16), 256, 0, stream>>>(x, conv_w, conv_b, what,
                                              out_xp, Z, rinv);
  for (int it = 0; it < 3; ++it) {
    k_u<<<B_ * M_, 256, 0, stream>>>(Z, vd, u);
    k_vdual<<<B_ * HW_ / 256, 256, 0, stream>>>(Z, u, vd);
  }
  k_vtilde<<<B_ * M_, 256, 0, stream>>>(Z, u, vd, out_xp, rinv, out_vt);
}